// RNN_5617817224124
// MI455X (gfx1250) — compile-verified
//
#include <hip/hip_runtime.h>

typedef float v2f __attribute__((ext_vector_type(2)));
typedef float v8f __attribute__((ext_vector_type(8)));

#define T_SEQ 2048
#define FEAT  256
#define GATES 64
#define HID   16
#define BATCH 64

// ---------------------------------------------------------------------------
// Kernel 1: fp32 WMMA GEMM, hoisted input-to-hidden projection of layer 0.
//   xg[b][g][t] = sum_k Wih0[g][k] * x[b][k][t] + bih0[g] + bhh0[g]
// One wave computes a 16(gate)x16(t) fp32 tile, K looped in steps of 4 via
// v_wmma_f32_16x16x4_f32. Block = 128 threads = 4 waves = 4 gate-tiles.
// Bandwidth-bound (134 MB of x @ 23.3 TB/s): fp32 matrix pipe is free here.
// ---------------------------------------------------------------------------
__global__ __launch_bounds__(128) void lstm_in_gemm_wmma(
    const float* __restrict__ x,     // [B][FEAT][T]
    const float* __restrict__ W,     // [GATES][FEAT]
    const float* __restrict__ bih,   // [GATES]
    const float* __restrict__ bhh,   // [GATES]
    float* __restrict__ xg)          // [B][GATES][T]
{
  const int lane = threadIdx.x & 31;
  const int m    = threadIdx.x >> 5;      // gate tile 0..3 (one per wave)
  const int half = lane >> 4;             // 0: lanes 0-15, 1: lanes 16-31
  const int r    = lane & 15;
  const int tb   = blockIdx.x * 16;       // t tile base
  const int b    = blockIdx.y;            // batch element

  const float* Xb   = x + (size_t)b * FEAT * T_SEQ;
  const float* Wrow = W + (size_t)(m * 16 + r) * FEAT;

  v8f acc = {};
  for (int k = 0; k < FEAT; k += 4) {
    // A (16x4 f32): lane = M row; VGPR0 holds K = k+2*half, VGPR1 holds K+1.
    const int ka = k + 2 * half;
    v2f av, bv;
    av.x = Wrow[ka];
    av.y = Wrow[ka + 1];
    // B (4x16 f32): VGPR0 row = k+2*half, VGPR1 row = +1; N = r (lane % 16).
    bv.x = Xb[(size_t)ka * T_SEQ + tb + r];
    bv.y = Xb[(size_t)(ka + 1) * T_SEQ + tb + r];
    acc = __builtin_amdgcn_wmma_f32_16x16x4_f32(
        /*neg_a=*/false, av, /*neg_b=*/false, bv,
        /*c_mod=*/(short)0, acc, /*reuse_a=*/false, /*reuse_b=*/false);
  }

  // C/D layout: VGPR v -> M = v (lanes 0-15) or v+8 (lanes 16-31), N = r.
#pragma unroll
  for (int v = 0; v < 8; ++v) {
    const int g = m * 16 + v + 8 * half;
    const float val = acc[v] + bih[g] + bhh[g];
    xg[((size_t)b * GATES + g) * T_SEQ + tb + r] = val;
  }
}

// ---------------------------------------------------------------------------
// Kernel 2: fused two-layer LSTM recurrence + hidden-mean.
// One wave per batch element. Lane l owns gates l and l+32 (torch order
// i=0..15, f=16..31, g=32..47, o=48..63), so for hidden unit j:
//   i[j]=lane j acc0, f[j]=lane j+16 acc0, g[j]=lane j acc1, o[j]=lane j+16 acc1
// Hidden state broadcasts use v_readlane (uniform index -> SGPR, no LDS pipe);
// only the half-swap gate gather needs a real permute (__shfl_xor 16).
// The h1 broadcast doubles as the hidden-mean reduction (uniform add tree).
// ---------------------------------------------------------------------------
__device__ __forceinline__ float bcast(float v, int j) {
  return __uint_as_float(__builtin_amdgcn_readlane(__float_as_uint(v), (unsigned)j));
}
__device__ __forceinline__ float sig_f(float x) {
  return __builtin_amdgcn_rcpf(1.0f + __expf(-x));
}
__device__ __forceinline__ float tanh_f(float x) {
  // saturating form with exact +-1 limits
  return 1.0f - 2.0f * __builtin_amdgcn_rcpf(__expf(2.0f * x) + 1.0f);
}

__global__ __launch_bounds__(32) void lstm_recurrence(
    const float* __restrict__ xg,    // [B][GATES][T] (layer-0 gates, biased)
    const float* __restrict__ Whh0,  // [GATES][HID]
    const float* __restrict__ Wih1,  // [GATES][HID]
    const float* __restrict__ Whh1,  // [GATES][HID]
    const float* __restrict__ bih1,  // [GATES]
    const float* __restrict__ bhh1,  // [GATES]
    float* __restrict__ out)         // [B][T]
{
  const int lane = threadIdx.x;
  const int b    = blockIdx.x;
  const int g0   = lane;
  const int g1   = lane + 32;

  // Per-lane weight rows resident in VGPRs for the whole sequence.
  float w0a[HID], w0b[HID], wia[HID], wib[HID], wha[HID], whb[HID];
#pragma unroll
  for (int j = 0; j < HID; ++j) {
    w0a[j] = Whh0[g0 * HID + j];
    w0b[j] = Whh0[g1 * HID + j];
    wia[j] = Wih1[g0 * HID + j];
    wib[j] = Wih1[g1 * HID + j];
    wha[j] = Whh1[g0 * HID + j];
    whb[j] = Whh1[g1 * HID + j];
  }
  const float bias0 = bih1[g0] + bhh1[g0];
  const float bias1 = bih1[g1] + bhh1[g1];

  const float* xg0p = xg + ((size_t)b * GATES + g0) * T_SEQ;
  const float* xg1p = xg + ((size_t)b * GATES + g1) * T_SEQ;
  float* outb = out + (size_t)b * T_SEQ;

  float h0 = 0.0f, c0 = 0.0f;   // lane j (<16) holds unit j state
  float h1 = 0.0f, c1 = 0.0f;
  float hb0[HID], hb1[HID];     // wave-uniform broadcast copies (SGPRs)
#pragma unroll
  for (int j = 0; j < HID; ++j) { hb0[j] = 0.0f; hb1[j] = 0.0f; }

  for (int t0 = 0; t0 < T_SEQ; t0 += 4) {
    const float4 x0v = *(const float4*)(xg0p + t0);
    const float4 x1v = *(const float4*)(xg1p + t0);
    __builtin_prefetch(xg0p + t0 + 64, 0, 3);
    __builtin_prefetch(xg1p + t0 + 64, 0, 3);
    const float x0a[4] = {x0v.x, x0v.y, x0v.z, x0v.w};
    const float x1a[4] = {x1v.x, x1v.y, x1v.z, x1v.w};

#pragma unroll
    for (int u = 0; u < 4; ++u) {
      // ---- layer 0: gates = xg0[t] + Whh0 @ h0 (hb0 = uniform h0) --------
      float p0a = 0.f, p0b = 0.f, p1a = 0.f, p1b = 0.f;
#pragma unroll
      for (int j = 0; j < 8; ++j) {
        p0a = fmaf(w0a[j],     hb0[j],     p0a);
        p0b = fmaf(w0a[j + 8], hb0[j + 8], p0b);
        p1a = fmaf(w0b[j],     hb0[j],     p1a);
        p1b = fmaf(w0b[j + 8], hb0[j + 8], p1b);
      }
      const float a0 = x0a[u] + (p0a + p0b);
      const float a1 = x1a[u] + (p1a + p1b);
      const float fv = __shfl_xor(a0, 16, 32);   // f[j] from lane j+16
      const float ov = __shfl_xor(a1, 16, 32);   // o[j] from lane j+16
      c0 = sig_f(fv) * c0 + sig_f(a0) * tanh_f(a1);
      h0 = sig_f(ov) * tanh_f(c0);

      // refresh uniform copy of new h0 (serves layer 1 now + layer 0 next t)
#pragma unroll
      for (int j = 0; j < HID; ++j) hb0[j] = bcast(h0, j);

      // ---- layer 1: gates = bias + Wih1 @ h0 + Whh1 @ h1_prev ------------
      float r0a = 0.f, r0b = 0.f, r0c = 0.f, r0d = 0.f;
      float r1a = 0.f, r1b = 0.f, r1c = 0.f, r1d = 0.f;
#pragma unroll
      for (int j = 0; j < 8; ++j) {
        r0a = fmaf(wia[j],     hb0[j],     r0a);
        r0b = fmaf(wia[j + 8], hb0[j + 8], r0b);
        r0c = fmaf(wha[j],     hb1[j],     r0c);
        r0d = fmaf(wha[j + 8], hb1[j + 8], r0d);
        r1a = fmaf(wib[j],     hb0[j],     r1a);
        r1b = fmaf(wib[j + 8], hb0[j + 8], r1b);
        r1c = fmaf(whb[j],     hb1[j],     r1c);
        r1d = fmaf(whb[j + 8], hb1[j + 8], r1d);
      }
      const float b0 = bias0 + ((r0a + r0b) + (r0c + r0d));
      const float b1 = bias1 + ((r1a + r1b) + (r1c + r1d));
      const float fv1 = __shfl_xor(b0, 16, 32);
      const float ov1 = __shfl_xor(b1, 16, 32);
      c1 = sig_f(fv1) * c1 + sig_f(b0) * tanh_f(b1);
      h1 = sig_f(ov1) * tanh_f(c1);

      // refresh uniform copy of new h1; mean falls out as a uniform tree sum
#pragma unroll
      for (int j = 0; j < HID; ++j) hb1[j] = bcast(h1, j);
      const float sA = (hb1[0] + hb1[1]) + (hb1[2] + hb1[3]);
      const float sB = (hb1[4] + hb1[5]) + (hb1[6] + hb1[7]);
      const float sC = (hb1[8] + hb1[9]) + (hb1[10] + hb1[11]);
      const float sD = (hb1[12] + hb1[13]) + (hb1[14] + hb1[15]);
      const float mean = ((sA + sB) + (sC + sD)) * 0.0625f;
      if (lane == 0) outb[t0 + u] = mean;
    }
  }
}

// ---------------------------------------------------------------------------
extern "C" void kernel_launch(void* const* d_in, const int* in_sizes, int n_in,
                              void* d_out, int out_size, void* d_ws, size_t ws_size,
                              hipStream_t stream) {
  (void)in_sizes; (void)n_in; (void)out_size; (void)ws_size;
  const float* x    = (const float*)d_in[0];
  const float* Wih0 = (const float*)d_in[1];
  const float* Whh0 = (const float*)d_in[2];
  const float* bih0 = (const float*)d_in[3];
  const float* bhh0 = (const float*)d_in[4];
  const float* Wih1 = (const float*)d_in[5];
  const float* Whh1 = (const float*)d_in[6];
  const float* bih1 = (const float*)d_in[7];
  const float* bhh1 = (const float*)d_in[8];
  float* out = (float*)d_out;
  float* xg  = (float*)d_ws;   // [B][GATES][T] = 33.5 MB scratch (L2-resident)

  // Kernel 1: WMMA fp32 GEMM over all (b, gate-tile, t-tile)
  dim3 g1(T_SEQ / 16, BATCH, 1);
  lstm_in_gemm_wmma<<<g1, dim3(128, 1, 1), 0, stream>>>(x, Wih0, bih0, bhh0, xg);

  // Kernel 2: one wave per batch element, fused 2-layer recurrence + mean
  lstm_recurrence<<<dim3(BATCH, 1, 1), dim3(32, 1, 1), 0, stream>>>(
      xg, Whh0, Wih1, Whh1, bih1, bhh1, out);
}